// RoundRobinGate_72980084293931
// MI455X (gfx1250) — compile-verified
//
#include <hip/hip_runtime.h>

// Round-robin MoE dispatch-mask builder for MI455X (gfx1250).
// Pure store-bandwidth problem: 537 MB output, no input reads, ~23us roofline
// at 23.3 TB/s. Only 16384 elements are 1.0 (one per row R with
// (R&15)==((R>>4)&15), at column (R>>8)&255); everything else is 0.
// Strategy: zero-fill a 32KB LDS stage (ds_store_b128 of a zero reg),
// sparsely poke the rare 1.0s, then drain with the CDNA5 async data mover
// (GLOBAL_STORE_ASYNC_FROM_LDS_B128, ASYNCcnt-tracked, NT hint to bypass
// the 192MB L2 for a 537MB stream). S_ENDPGM's implicit wait-idle drains
// ASYNCcnt before LDS is freed.

#define TPB 256
#define CHUNK_F4 2048                 // float4 slots staged in LDS = 32 KB
#define CHUNK_FLOATS (CHUNK_F4 * 4)   // 8192 floats per block

typedef float __attribute__((ext_vector_type(4))) f4;

// Async LDS->global store, 16B/lane (512B/wave). INST_OFFSET advances both
// the LDS and global addresses in lockstep (ISA 15.18.3), so one base pair
// serves all 8 issues.
template <int OFF>
__device__ __forceinline__ void async_store_b128(unsigned long long gaddr,
                                                 unsigned lds_off) {
  asm volatile(
      "global_store_async_from_lds_b128 %0, %1, off offset:%2 th:TH_STORE_NT"
      :: "v"(gaddr), "v"(lds_off), "n"(OFF)
      : "memory");
}

// Scalar fallback for the single tail element (out_elems % 4 == 1).
__device__ __forceinline__ float val_at(unsigned pos, unsigned n_region) {
  if (pos == 0) return 0.0f;                 // leading scalar output (0.0)
  unsigned q = pos - 1;
  if (q >= n_region) q -= n_region;          // bool region mirrors float region
  unsigned R = q >> 9;                       // row = (g<<16)|(i<<4)|e
  unsigned c = q & 511u;                     // capacity slot
  return (((R & 15u) == ((R >> 4) & 15u)) && (c == ((R >> 8) & 255u)))
             ? 1.0f : 0.0f;
}

__global__ void __launch_bounds__(TPB)
RoundRobinGate_fill(float* __restrict__ out, unsigned out_elems) {
  __shared__ f4 stage[CHUNK_F4];
  const unsigned n_region = (out_elems - 1) >> 1;   // 2^26 elements per region
  const unsigned m4 = out_elems >> 2;               // full float4 count
  const unsigned P  = blockIdx.x * CHUNK_FLOATS;    // first output float of chunk
  const unsigned t  = threadIdx.x;

  // 1) Zero-fill the stage: 8x ds_store_b128 of a zero register per thread.
  const f4 z = {0.0f, 0.0f, 0.0f, 0.0f};
#pragma unroll
  for (unsigned u = 0; u < CHUNK_F4 / TPB; ++u)
    stage[u * TPB + t] = z;
  __syncthreads();

  // 2) Sparse poke: at most 17 candidate rows overlap this chunk per region.
  //    Row R carries a 1 at column c=(R>>8)&255 iff (R&15)==((R>>4)&15).
  if (t < 20u) {
#pragma unroll
    for (int reg = 0; reg < 2; ++reg) {
      const long long Pq = (long long)P - 1 - (long long)reg * n_region;
      const long long R  = (Pq >> 9) + (long long)t;   // arithmetic shift floors
      if (R >= 0 && R < (long long)(n_region >> 9)) {
        const unsigned Ru = (unsigned)R;
        if ((Ru & 15u) == ((Ru >> 4) & 15u)) {
          const unsigned c = (Ru >> 8) & 255u;
          const long long rel = (((long long)Ru << 9) + c) - Pq;  // pos - P
          if (rel >= 0 && rel < (long long)CHUNK_FLOATS)
            ((float*)stage)[rel] = 1.0f;
        }
      }
    }
  }
  __syncthreads();

  // 3) Drain LDS -> global with the async data mover (wave-contiguous 512B).
  const unsigned base4 = blockIdx.x * CHUNK_F4;
  const unsigned long long gaddr =
      (unsigned long long)out + (unsigned long long)(base4 + t) * 16ull;
  const unsigned lds_off = (unsigned)(unsigned long long)&stage[t];

  if (base4 + CHUNK_F4 <= m4) {        // full chunk (always true for this shape)
    async_store_b128<0 * 4096>(gaddr, lds_off);
    async_store_b128<1 * 4096>(gaddr, lds_off);
    async_store_b128<2 * 4096>(gaddr, lds_off);
    async_store_b128<3 * 4096>(gaddr, lds_off);
    async_store_b128<4 * 4096>(gaddr, lds_off);
    async_store_b128<5 * 4096>(gaddr, lds_off);
    async_store_b128<6 * 4096>(gaddr, lds_off);
    async_store_b128<7 * 4096>(gaddr, lds_off);
  } else {                             // partial last chunk (generic sizes)
#pragma unroll
    for (unsigned u = 0; u < CHUNK_F4 / TPB; ++u) {
      if (base4 + u * TPB + t < m4)
        async_store_b128<0>(gaddr + u * 4096ull, lds_off + u * 4096u);
    }
  }

  // 4) Tail element(s): out_elems = 4*m4 + 1 -> one plain store.
  if (blockIdx.x == 0 && t == 0) {
    for (unsigned pos = m4 * 4u; pos < out_elems; ++pos)
      out[pos] = val_at(pos, n_region);
  }
}

extern "C" void kernel_launch(void* const* d_in, const int* in_sizes, int n_in,
                              void* d_out, int out_size, void* d_ws, size_t ws_size,
                              hipStream_t stream) {
  (void)d_in; (void)in_sizes; (void)n_in; (void)d_ws; (void)ws_size;
  float* out = (float*)d_out;
  const unsigned out_elems = (unsigned)out_size;
  const unsigned m4 = out_elems >> 2;
  unsigned blocks = (m4 + CHUNK_F4 - 1) / CHUNK_F4;
  if (blocks == 0) blocks = 1;
  RoundRobinGate_fill<<<blocks, TPB, 0, stream>>>(out, out_elems);
}